// DetectionLoss_26371099197480
// MI455X (gfx1250) — compile-verified
//
#include <hip/hip_runtime.h>
#include <math.h>

#define NB   1196
#define NT   64
#define BATCH 256

typedef __attribute__((ext_vector_type(2))) float v2f;
typedef __attribute__((ext_vector_type(8))) float v8f;
typedef __attribute__((ext_vector_type(4))) unsigned int v4u;
typedef __attribute__((ext_vector_type(8))) int v8i;
typedef __attribute__((ext_vector_type(4))) int v4i;

// Reduce 64 f32 partials (in LDS) to one value using V_WMMA_F32_16X16X4_F32
// against an all-ones B matrix. Must be called by all 32 lanes of a wave
// (EXEC all ones). Valid result returned on lane 0.
// A-matrix 16x4 f32 layout (ISA 7.12.2): lanes 0-15 hold K=0,1 in VGPR0,1;
// lanes 16-31 hold K=2,3. s64 is indexed as [m*4 + k].
// D (16x16 f32): VGPR r, lane l -> M = r + 8*(l>=16), N = l%16. Column N=0
// lives in lanes 0 and 16; summing the 8 accumulator regs there gives the
// two half-sums.
__device__ __forceinline__ float wmma_reduce64(const float* __restrict__ s64,
                                               unsigned lane) {
    unsigned m  = lane & 15u;
    unsigned kh = (lane >> 4) << 1;
    v2f a;
    a[0] = s64[m * 4 + kh];
    a[1] = s64[m * 4 + kh + 1];
    v2f bo;
    bo[0] = 1.0f;
    bo[1] = 1.0f;
    v8f c = {};
    v8f d = __builtin_amdgcn_wmma_f32_16x16x4_f32(
        /*neg_a=*/false, a, /*neg_b=*/false, bo,
        /*c_mod=*/(short)0, c, /*reuse_a=*/false, /*reuse_b=*/false);
    float s = ((((((d[0] + d[1]) + d[2]) + d[3]) + d[4]) + d[5]) + d[6]) + d[7];
    float s_hi = __shfl(s, 16, 32);   // lane16 holds M=8..15 of column 0
    return s + s_hi;                  // meaningful on lane 0
}

__device__ __forceinline__ float smooth_l1(float d) {
    float ad = fabsf(d);
    return (ad < 1.0f) ? 0.5f * d * d : (ad - 0.5f);
}

__global__ __launch_bounds__(256) void det_loss_main(
    const float* __restrict__ pred,    // (B, N, 9)
    const float* __restrict__ tgt,     // (B, T, 5)
    float* __restrict__ partials)      // (B,)
{
    __shared__ float bx1[NB], by1[NB], bx2[NB], by2[NB], ba[NB];
    __shared__ float stgt[NT * 5];
    __shared__ float bestv[256];
    __shared__ int   besti[256];
    __shared__ int   mi[NT];
    __shared__ unsigned posmask[(NB + 31) / 32];
    __shared__ float red[256];
    __shared__ float s64[64];

    const unsigned tid = threadIdx.x;
    const unsigned b   = blockIdx.x;
    const float* pb = pred + (size_t)b * NB * 9;

    if (tid < (NB + 31) / 32) posmask[tid] = 0u;

    // ---- TDM: DMA targets[b] (64x5 f32 = 320 dwords, contiguous) into LDS ----
    if (tid == 0) {
        unsigned long long ga =
            (unsigned long long)(size_t)(const void*)(tgt + (size_t)b * NT * 5);
        unsigned ldsoff = (unsigned)(size_t)(void*)&stgt[0];
        const unsigned elems = NT * 5;   // 320 elements, 1-D tile
        v4u g0;
        g0[0] = 1u;                                          // count=1, user desc
        g0[1] = ldsoff;                                      // lds_addr
        g0[2] = (unsigned)(ga & 0xFFFFFFFFull);              // global_addr lo
        g0[3] = (unsigned)((ga >> 32) & 0x01FFFFFFull)       // global_addr hi
              | (2u << 30);                                  // type=2 (image)
        v8i g1;
        g1[0] = (int)(2u << 16);                             // data_size=4B
        g1[1] = (int)((elems & 0xFFFFu) << 16);              // tensor_dim0 lo16
        g1[2] = (int)((elems >> 16) | (1u << 16));           // dim0 hi16 | tensor_dim1=1
        g1[3] = (int)((elems & 0xFFFFu) << 16);              // dim1 hi16=0 | tile_dim0
        g1[4] = (int)1;                                      // tile_dim1=1, tile_dim2=0
        g1[5] = (int)elems;                                  // tensor_dim0_stride lo32
        g1[6] = (int)((elems & 0xFFFFu) << 16);              // stride0 hi=0 | stride1 lo16
        g1[7] = (int)(elems >> 16);                          // stride1 hi
        v4i gz4 = {0, 0, 0, 0};
        v8i gz8 = {0, 0, 0, 0, 0, 0, 0, 0};
        __builtin_amdgcn_tensor_load_to_lds(g0, g1, gz4, gz4, gz8, 0);
    }
    __builtin_amdgcn_s_wait_tensorcnt(0);

    // ---- decode boxes into LDS (sx = 1472/46 = 32, sy = 832/26 = 32) ----
    for (unsigned n = tid; n < NB; n += 256) {
        const float* pp = pb + (size_t)n * 9;
        float cx = (pp[0] * 2.0f - 1.0f) * 736.0f;
        float cy = (pp[1] * 2.0f - 1.0f) * 416.0f;
        float bw = expf(pp[2]) * 32.0f;
        float bh = expf(pp[3]) * 32.0f;
        float x1 = cx - bw * 0.5f, x2 = cx + bw * 0.5f;
        float y1 = cy - bh * 0.5f, y2 = cy + bh * 0.5f;
        bx1[n] = x1; by1[n] = y1; bx2[n] = x2; by2[n] = y2;
        ba[n] = (x2 - x1) * (y2 - y1);
    }
    __syncthreads();

    // ---- matching: 4 threads per target, first-max-index semantics ----
    {
        const unsigned t = tid >> 2, r = tid & 3;
        const float tx1 = stgt[t * 5 + 0], ty1 = stgt[t * 5 + 1];
        const float tx2 = stgt[t * 5 + 2], ty2 = stgt[t * 5 + 3];
        const float ta = (tx2 - tx1) * (ty2 - ty1);
        float bv = -1.0f; int bi = 0;
        for (unsigned n = r; n < NB; n += 4) {
            float iw = fminf(bx2[n], tx2) - fmaxf(bx1[n], tx1);
            float ih = fminf(by2[n], ty2) - fmaxf(by1[n], ty1);
            iw = fmaxf(iw, 0.0f); ih = fmaxf(ih, 0.0f);
            float inter = iw * ih;
            float uni = ba[n] + ta - inter;
            float iou = (uni > 0.0f) ? inter / fmaxf(uni, 1e-12f) : 0.0f;
            if (iou > bv) { bv = iou; bi = (int)n; }   // strict > : first index
        }
        bestv[tid] = bv; besti[tid] = bi;
    }
    __syncthreads();
    if (tid < NT) {
        float v = bestv[tid * 4]; int i = besti[tid * 4];
#pragma unroll
        for (int r = 1; r < 4; ++r) {
            float v2 = bestv[tid * 4 + r]; int i2 = besti[tid * 4 + r];
            if (v2 > v || (v2 == v && i2 < i)) { v = v2; i = i2; }  // tie -> lower n
        }
        mi[tid] = i;
        atomicOr(&posmask[i >> 5], 1u << (i & 31));
    }
    __syncthreads();

    // ---- per-target box (smooth L1) + class (CE) losses ----
    float part = 0.0f;
    if (tid < NT) {
        const unsigned t = tid;
        const int m = mi[t];
        const float tx1 = stgt[t * 5 + 0], ty1 = stgt[t * 5 + 1];
        const float tx2 = stgt[t * 5 + 2], ty2 = stgt[t * 5 + 3];
        float bl = smooth_l1(bx1[m] - tx1) + smooth_l1(by1[m] - ty1)
                 + smooth_l1(bx2[m] - tx2) + smooth_l1(by2[m] - ty2);
        const float* lp = pb + (size_t)m * 9 + 5;
        float l0 = lp[0], l1 = lp[1], l2 = lp[2], l3 = lp[3];
        float mx = fmaxf(fmaxf(l0, l1), fmaxf(l2, l3));
        float lse = mx + logf(expf(l0 - mx) + expf(l1 - mx) +
                              expf(l2 - mx) + expf(l3 - mx));
        int lab = (int)stgt[t * 5 + 4];
        float ll = (lab == 0) ? l0 : (lab == 1) ? l1 : (lab == 2) ? l2 : l3;
        part += 5.0f * bl + (lse - ll);      // LAMBDA_BOX=5, LAMBDA_CLS=1
    }

    // ---- confidence BCE over all N ----
    for (unsigned n = tid; n < NB; n += 256) {
        float x = pb[(size_t)n * 9 + 4];
        float tp = (float)((posmask[n >> 5] >> (n & 31)) & 1u);
        part += fmaxf(x, 0.0f) - x * tp + log1pf(expf(-fabsf(x)));
    }

    // ---- deterministic block reduction: 256 -> 64 (LDS) -> 1 (WMMA) ----
    red[tid] = part;
    __syncthreads();
    if (tid < 64)
        s64[tid] = ((red[tid] + red[tid + 64]) + red[tid + 128]) + red[tid + 192];
    __syncthreads();
    if (tid < 32) {
        float total = wmma_reduce64(s64, tid);
        if (tid == 0) partials[b] = total;
    }
}

__global__ __launch_bounds__(64) void det_loss_final(
    const float* __restrict__ partials, float* __restrict__ out)
{
    __shared__ float s64[64];
    const unsigned tid = threadIdx.x;
    s64[tid] = ((partials[tid * 4] + partials[tid * 4 + 1]) +
                partials[tid * 4 + 2]) + partials[tid * 4 + 3];
    __syncthreads();
    if (tid < 32) {
        float total = wmma_reduce64(s64, tid);
        if (tid == 0) out[0] = total * (1.0f / (float)BATCH);
    }
}

extern "C" void kernel_launch(void* const* d_in, const int* in_sizes, int n_in,
                              void* d_out, int out_size, void* d_ws, size_t ws_size,
                              hipStream_t stream) {
    const float* pred = (const float*)d_in[0];   // (256, 1196, 9) f32
    const float* tgt  = (const float*)d_in[1];   // (256, 64, 5)  f32
    float* partials = (float*)d_ws;              // 256 floats of scratch
    det_loss_main<<<BATCH, 256, 0, stream>>>(pred, tgt, partials);
    det_loss_final<<<1, 64, 0, stream>>>(partials, (float*)d_out);
}